// QuantumPureBosonicExt_31619549233405
// MI455X (gfx1250) — compile-verified
//
#include <hip/hip_runtime.h>

// ---------------- complex helpers (fp64) ----------------
__device__ __forceinline__ double2 czero() { return make_double2(0.0, 0.0); }
__device__ __forceinline__ double2 cmul(double2 a, double2 b) {
  return make_double2(fma(a.x, b.x, -a.y * b.y), fma(a.x, b.y, a.y * b.x));
}
__device__ __forceinline__ void cfma(double2& acc, double2 a, double2 b) {
  acc.x = fma(a.x, b.x, fma(-a.y, b.y, acc.x));
  acc.y = fma(a.x, b.y, fma(a.y, b.x, acc.y));
}
__device__ __forceinline__ double2 conjc(double2 a) { return make_double2(a.x, -a.y); }
__device__ __forceinline__ int imin(int a, int b) { return a < b ? a : b; }

#define TWO_PI 6.2831853071795864769252867665590057684

// ---------------- TDM plumbing (gfx1250) ----------------
typedef unsigned int u32x4 __attribute__((ext_vector_type(4)));
typedef int          i32x8 __attribute__((ext_vector_type(8)));
typedef int          i32x4 __attribute__((ext_vector_type(4)));

#if defined(__has_builtin)
#if __has_builtin(__builtin_amdgcn_tensor_load_to_lds)
#define KD_USE_TDM 1
#endif
#endif

__device__ __forceinline__ void wait_tensorcnt0() {
#if defined(__has_builtin) && __has_builtin(__builtin_amdgcn_s_wait_tensorcnt)
  __builtin_amdgcn_s_wait_tensorcnt(0);
#else
  asm volatile("s_wait_tensorcnt 0x0" ::: "memory");
#endif
}

// 1-D contiguous TDM copy: `elems` 8-byte words from global `gaddr` to LDS `ldsoff`.
// D# packing per CDNA5 ISA 8.3/8.4.
__device__ __forceinline__ void tdm_load_1d(unsigned ldsoff, unsigned long long gaddr,
                                            unsigned elems) {
#if defined(KD_USE_TDM)
  u32x4 g0;
  g0[0] = 1u;                                   // count=1, user descriptor
  g0[1] = ldsoff;                               // lds_addr (bytes)
  g0[2] = (unsigned)(gaddr & 0xffffffffull);    // global_addr[31:0]
  g0[3] = (unsigned)((gaddr >> 32) & 0x1ffffffull) | (2u << 30);  // addr[56:32] | type=2
  i32x8 g1;
  g1[0] = 0x00030000;                           // data_size = 3 (8 bytes)
  g1[1] = (int)((elems & 0xffffu) << 16);       // tensor_dim0[15:0]
  g1[2] = (int)(((elems >> 16) & 0xffffu) | (1u << 16));  // dim0[31:16] | tensor_dim1=1
  g1[3] = (int)((elems & 0xffffu) << 16);       // tile_dim0 (<= 65535)
  g1[4] = 0;                                    // tile_dim1 = 0 (unused)
  g1[5] = (int)elems;                           // tensor_dim0_stride[31:0]
  g1[6] = 0;
  g1[7] = 0;
  i32x4 g2 = {0, 0, 0, 0};
  i32x4 g3 = {0, 0, 0, 0};
#if defined(__clang_major__) && (__clang_major__ >= 23)
  i32x8 gz = {0, 0, 0, 0, 0, 0, 0, 0};
  __builtin_amdgcn_tensor_load_to_lds(g0, g1, g2, g3, gz, 0);
#else
  __builtin_amdgcn_tensor_load_to_lds(g0, g1, g2, g3, 0);
#endif
  wait_tensorcnt0();
#else
  (void)ldsoff; (void)gaddr; (void)elems;
#endif
}

// Per-lane async gather: 16 bytes global -> LDS, tracked by ASYNCcnt.
__device__ __forceinline__ void async_gather_b128(unsigned lds_off, const void* gsrc) {
  unsigned long long ga = (unsigned long long)(size_t)gsrc;
  asm volatile("global_load_async_to_lds_b128 %0, %1, off"
               :: "v"(lds_off), "v"(ga) : "memory");
}
__device__ __forceinline__ void wait_asynccnt0() {
  asm volatile("s_wait_asynccnt 0x0" ::: "memory");
}

// ---------------- K0: build the layer unitaries ----------------
__global__ void __launch_bounds__(32)
k_unitaries(const double* __restrict__ theta, const double2* __restrict__ H,
            const double2* __restrict__ mps_p0, double2* __restrict__ uab,
            double2* __restrict__ mps_p, int L, int NM) {
  int w = threadIdx.x;
  if (w >= 2 * L) return;
  int l = w >> 1, s = w & 1;

  double2 h[9], hc[9];
  for (int i = 0; i < 9; ++i) h[i] = H[i];
  for (int r = 0; r < 3; ++r)
    for (int c = 0; c < 3; ++c) hc[r * 3 + c] = conjc(H[c * 3 + r]);

  double2 U[9], M[9], T[9];
  for (int i = 0; i < 4; ++i) {
    double tx = theta[((l * 2 + s) * 4 + i) * 2 + 0];
    double tz = theta[((l * 2 + s) * 4 + i) * 2 + 1];
    double2 X[3], Z[3];
    for (int d = 0; d < 3; ++d) {
      double sn, cs;
      sincos(TWO_PI * (double)d * tx, &sn, &cs);
      X[d] = make_double2(cs, sn);
      sincos(TWO_PI * (double)d * tz, &sn, &cs);
      Z[d] = make_double2(cs, sn);
    }
    for (int r = 0; r < 3; ++r)
      for (int c = 0; c < 3; ++c) {
        double2 acc = czero();
        for (int k = 0; k < 3; ++k) {
          double2 t = cmul(h[r * 3 + k], X[k]);
          cfma(acc, t, hc[k * 3 + c]);
        }
        M[r * 3 + c] = cmul(acc, Z[c]);
      }
    if (i == 0) {
      for (int q = 0; q < 9; ++q) U[q] = M[q];
    } else {
      for (int r = 0; r < 3; ++r)
        for (int c = 0; c < 3; ++c) {
          double2 acc = czero();
          for (int k = 0; k < 3; ++k) cfma(acc, M[r * 3 + k], U[k * 3 + c]);
          T[r * 3 + c] = acc;
        }
      for (int q = 0; q < 9; ++q) U[q] = T[q];
    }
  }
  double2* dst = uab + (size_t)(l * 2 + s) * 9;
  for (int q = 0; q < 9; ++q) dst[q] = U[q];

  if (l == 0 && s == 0) {  // mps_p(layer0) = UA0 @ mps_p0 (3x1)
    for (int a = 0; a < 3; ++a) {
      double2 acc = czero();
      for (int b = 0; b < 3; ++b) cfma(acc, U[a * 3 + b], mps_p0[b]);
      mps_p[(size_t)a * NM] = acc;
    }
  }
}

// ---------------- K1: alpha2 = abase @ UB_l^T ; power tables [m][d][0..56] -------
__global__ void __launch_bounds__(256)
k_powtab(const double2* __restrict__ abase, const double2* __restrict__ uab, int l,
         int r, double2* __restrict__ powtab) {
  int t = blockIdx.x * blockDim.x + threadIdx.x;
  if (t >= 3 * r) return;
  int m = t / 3, d = t - 3 * m;
  const double2* UB = uab + (size_t)(l * 2 + 1) * 9;
  double2 a = czero();
  for (int e = 0; e < 3; ++e) cfma(a, abase[(size_t)m * 3 + e], UB[d * 3 + e]);
  double2* row = powtab + ((size_t)m * 3 + d) * 57;
  double2 pw = make_double2(1.0, 0.0);
  for (int p = 0; p < 57; ++p) { row[p] = pw; pw = cmul(pw, a); }
}

// ---- K2a: partial dicke over an m-slice: dpart[slice][a][j] = sum_m p[a][m]*PW[m][j]
#define TM 16  // pow-table rows staged per LDS tile (16*171*16B = 43.7 KB)
__global__ void __launch_bounds__(256)
k_dicke_part(const double2* __restrict__ powtab, const double2* __restrict__ mps_p,
             const long long* __restrict__ klist, double2* __restrict__ dpart,
             int r, int ND, int NM, int MS) {
  __shared__ double2 tilePow[TM * 171];
  __shared__ double2 tileP[3 * TM];
  int tid = threadIdx.x;
  int j = blockIdx.x * blockDim.x + tid;
  int slice = blockIdx.y;
  int per = (r + MS - 1) / MS;
  int mb = slice * per;
  int me = imin(r, mb + per);

  int k0 = 0, k1 = 0, k2 = 0;
  if (j < ND) {
    k0 = (int)klist[(size_t)j * 3 + 0];
    k1 = (int)klist[(size_t)j * 3 + 1];
    k2 = (int)klist[(size_t)j * 3 + 2];
  }
  double2 acc0 = czero(), acc1 = czero(), acc2 = czero();

  for (int m0 = mb; m0 < me; m0 += TM) {
    int tm = imin(TM, me - m0);
    __syncthreads();  // previous tile fully consumed
#if defined(KD_USE_TDM)
    if (tid < 32) {   // one wave issues the TDM descriptor (EXEC-independent DMA)
      unsigned long long gaddr =
          (unsigned long long)(size_t)(powtab + (size_t)m0 * 171);
      unsigned ldsoff = (unsigned)(size_t)(void*)tilePow;
      tdm_load_1d(ldsoff, gaddr, (unsigned)(tm * 342));
    }
#else
    for (int idx = tid; idx < tm * 171; idx += blockDim.x)
      tilePow[idx] = powtab[(size_t)m0 * 171 + idx];
#endif
    if (tid < 3 * TM) {
      int a = tid / TM, i = tid % TM;
      tileP[tid] = (i < tm) ? mps_p[(size_t)a * NM + m0 + i] : czero();
    }
    __syncthreads();
    if (j < ND) {
      for (int i = 0; i < tm; ++i) {
        double2 q = cmul(tilePow[i * 171 + k0], tilePow[i * 171 + 57 + k1]);
        q = cmul(q, tilePow[i * 171 + 114 + k2]);
        cfma(acc0, tileP[0 * TM + i], q);
        cfma(acc1, tileP[1 * TM + i], q);
        cfma(acc2, tileP[2 * TM + i], q);
      }
    }
  }
  if (j < ND) {
    dpart[((size_t)slice * 3 + 0) * ND + j] = acc0;
    dpart[((size_t)slice * 3 + 1) * ND + j] = acc1;
    dpart[((size_t)slice * 3 + 2) * ND + j] = acc2;
  }
}

// ---- K2b: dicke[a][j] = binom[j] * sum_slices dpart
__global__ void __launch_bounds__(256)
k_dicke_combine(const double2* __restrict__ dpart, const double* __restrict__ binom,
                double2* __restrict__ dicke, int ND, int MS) {
  int t = blockIdx.x * blockDim.x + threadIdx.x;
  if (t >= 3 * ND) return;
  int a = t / ND, j = t - a * ND;
  double2 s = czero();
  for (int sl = 0; sl < MS; ++sl) {
    double2 v = dpart[((size_t)sl * 3 + a) * ND + j];
    s.x += v.x; s.y += v.y;
  }
  double b = binom[j];
  dicke[(size_t)a * ND + j] = make_double2(s.x * b, s.y * b);
}

// ---- K3a: partial gemm over a j-slice: gpart[slice][a][n] = sum_j dperm[a][j]*matB[j][n]
#define TJ 64
__global__ void __launch_bounds__(256)
k_gemm_part(const double2* __restrict__ dicke, const long long* __restrict__ perm,
            const double2* __restrict__ matB, double2* __restrict__ gpart,
            int ND, int NM, int JS) {
  __shared__ double2 dperm[3 * TJ];
  int tid = threadIdx.x;
  int n = blockIdx.x * blockDim.x + tid;
  int slice = blockIdx.y;
  int per = (ND + JS - 1) / JS;
  int jb = slice * per;
  int je = imin(ND, jb + per);
  double2 acc0 = czero(), acc1 = czero(), acc2 = czero();

  for (int j0 = jb; j0 < je; j0 += TJ) {
    int tj = imin(TJ, je - j0);
    __syncthreads();
    if (tid < 3 * TJ) {  // waves 0..5: per-lane async gather global -> LDS
      int a = tid / TJ, jj = tid % TJ;
      if (jj < tj) {
        const double2* src =
            &dicke[(size_t)a * ND + (size_t)perm[(size_t)a * ND + j0 + jj]];
        async_gather_b128((unsigned)(size_t)(&dperm[tid]), src);
      } else {
        dperm[tid] = czero();
      }
    }
    wait_asynccnt0();
    __syncthreads();
    if (n < NM) {
      if (j0 + TJ < je)  // stream matB through L2 -> global_prefetch_b8
        __builtin_prefetch(&matB[(size_t)(j0 + TJ) * NM + n], 0, 1);
      for (int jj = 0; jj < tj; ++jj) {
        double2 b = matB[(size_t)(j0 + jj) * NM + n];
        cfma(acc0, dperm[0 * TJ + jj], b);
        cfma(acc1, dperm[1 * TJ + jj], b);
        cfma(acc2, dperm[2 * TJ + jj], b);
      }
    }
  }
  if (n < NM) {
    gpart[((size_t)slice * 3 + 0) * NM + n] = acc0;
    gpart[((size_t)slice * 3 + 1) * NM + n] = acc1;
    gpart[((size_t)slice * 3 + 2) * NM + n] = acc2;
  }
}

// ---- K3b: mps_p[b][n] = sum_a UA_next[b][a] * sum_slices gpart[slice][a][n]
__global__ void __launch_bounds__(256)
k_gemm_combine(const double2* __restrict__ gpart, const double2* __restrict__ uaNext,
               double2* __restrict__ mps_p, int NM, int JS) {
  int n = blockIdx.x * blockDim.x + threadIdx.x;
  if (n >= NM) return;
  double2 A0 = czero(), A1 = czero(), A2 = czero();
  for (int sl = 0; sl < JS; ++sl) {
    double2 v0 = gpart[((size_t)sl * 3 + 0) * NM + n];
    double2 v1 = gpart[((size_t)sl * 3 + 1) * NM + n];
    double2 v2 = gpart[((size_t)sl * 3 + 2) * NM + n];
    A0.x += v0.x; A0.y += v0.y;
    A1.x += v1.x; A1.y += v1.y;
    A2.x += v2.x; A2.y += v2.y;
  }
  double2 U[9];
  for (int q = 0; q < 9; ++q) U[q] = uaNext[q];
  for (int b = 0; b < 3; ++b) {
    double2 o = czero();
    cfma(o, U[b * 3 + 0], A0);
    cfma(o, U[b * 3 + 1], A1);
    cfma(o, U[b * 3 + 2], A2);
    mps_p[(size_t)b * NM + n] = o;
  }
}

// ---------------- K4: final bilinear reduction -> Re(<rho, op>) ----------------
__global__ void __launch_bounds__(1024)
k_final(const double2* __restrict__ dicke, const long long* __restrict__ perm,
        const long long* __restrict__ pt_m, const long long* __restrict__ pt_n,
        const long long* __restrict__ pt_ij, const double* __restrict__ coef,
        const double2* __restrict__ opv, double* __restrict__ out, int ND, int NPT) {
  __shared__ double2 ops[81];
  __shared__ double red[1024];
  int tid = threadIdx.x;
  if (tid < 81) ops[tid] = opv[tid];
  __syncthreads();
  double s = 0.0;
  for (int t = tid; t < NPT; t += blockDim.x) {
    int m = (int)pt_m[t], n = (int)pt_n[t], ij = (int)pt_ij[t];
    double c = coef[t];
    int i = ij / 3, jj = ij - 3 * i;
    double2 dA[3], dBc[3];
    for (int a = 0; a < 3; ++a)
      dA[a] = dicke[(size_t)a * ND + (size_t)perm[(size_t)a * ND + m]];
    for (int b = 0; b < 3; ++b)
      dBc[b] = conjc(dicke[(size_t)b * ND + (size_t)perm[(size_t)b * ND + n]]);
    double part = 0.0;
    for (int a = 0; a < 3; ++a)
      for (int b = 0; b < 3; ++b) {
        double2 z = cmul(dA[a], dBc[b]);
        double2 w = ops[(a * 3 + i) * 9 + (b * 3 + jj)];
        part += z.x * w.x - z.y * w.y;  // Re(z*w)
      }
    s += c * part;
  }
  red[tid] = s;
  __syncthreads();
  for (int st = blockDim.x / 2; st > 0; st >>= 1) {
    if (tid < st) red[tid] += red[tid + st];
    __syncthreads();
  }
  if (tid == 0) out[0] = red[0];
}

// ---------------- host-side orchestration ----------------
static inline size_t al256(size_t x) { return (x + 255) & ~(size_t)255; }
#define MSLICES 14
#define JSLICES 13

extern "C" void kernel_launch(void* const* d_in, const int* in_sizes, int n_in,
                              void* d_out, int out_size, void* d_ws, size_t ws_size,
                              hipStream_t stream) {
  (void)n_in; (void)out_size; (void)ws_size;
  const double*    theta      = (const double*)d_in[0];
  const double2*   weylH      = (const double2*)d_in[1];
  const double2*   mps_p0     = (const double2*)d_in[2];
  const double2*   mps_alpha0 = (const double2*)d_in[3];
  const long long* klist      = (const long long*)d_in[4];
  const double*    binom      = (const double*)d_in[5];
  const long long* perm       = (const long long*)d_in[6];
  const double2*   matB       = (const double2*)d_in[7];
  const double2*   basis      = (const double2*)d_in[8];
  const long long* pt_m       = (const long long*)d_in[9];
  const long long* pt_n       = (const long long*)d_in[10];
  const long long* pt_ij      = (const long long*)d_in[11];
  const double*    pt_coef    = (const double*)d_in[12];
  const double2*   op_T_vec   = (const double2*)d_in[13];

  const int ND  = in_sizes[5];            // 1653
  const int L   = in_sizes[0] / 16;       // 8
  int NM        = in_sizes[8] / 3;        // 1735
  if (NM > 2 * ND) NM /= 2;
  const int NPT = in_sizes[9];

  char* ws = (char*)d_ws;
  size_t off = 0;
  double2* uab    = (double2*)(ws + off); off = al256(off + (size_t)L * 2 * 9 * 16);
  double2* mps_p  = (double2*)(ws + off); off = al256(off + (size_t)3 * NM * 16);
  double2* powtab = (double2*)(ws + off); off = al256(off + (size_t)NM * 3 * 57 * 16);
  double2* dicke  = (double2*)(ws + off); off = al256(off + (size_t)3 * ND * 16);
  double2* dpart  = (double2*)(ws + off); off = al256(off + (size_t)MSLICES * 3 * ND * 16);
  double2* gpart  = (double2*)(ws + off); off = al256(off + (size_t)JSLICES * 3 * NM * 16);

  k_unitaries<<<1, 32, 0, stream>>>(theta, weylH, mps_p0, uab, mps_p, L, NM);

  const int jblocks = (ND + 255) / 256;
  const int nblocks = (NM + 255) / 256;
  for (int l = 0; l < L; ++l) {
    const int r = (l == 0) ? 1 : NM;
    const int ms = (l == 0) ? 1 : MSLICES;
    const double2* abase = (l == 0) ? mps_alpha0 : basis;
    k_powtab<<<(3 * r + 255) / 256, 256, 0, stream>>>(abase, uab, l, r, powtab);
    k_dicke_part<<<dim3(jblocks, ms), 256, 0, stream>>>(powtab, mps_p, klist,
                                                        dpart, r, ND, NM, ms);
    k_dicke_combine<<<(3 * ND + 255) / 256, 256, 0, stream>>>(dpart, binom, dicke,
                                                              ND, ms);
    if (l + 1 < L) {
      k_gemm_part<<<dim3(nblocks, JSLICES), 256, 0, stream>>>(dicke, perm, matB,
                                                              gpart, ND, NM, JSLICES);
      k_gemm_combine<<<nblocks, 256, 0, stream>>>(gpart,
                                                  uab + (size_t)(l + 1) * 2 * 9,
                                                  mps_p, NM, JSLICES);
    }
  }
  k_final<<<1, 1024, 0, stream>>>(dicke, perm, pt_m, pt_n, pt_ij, pt_coef,
                                  op_T_vec, (double*)d_out, ND, NPT);
}